// TriangleMultiplicationCpp_62182536511497
// MI455X (gfx1250) — compile-verified
//
#include <hip/hip_runtime.h>
#include <math.h>

#define NN 512
#define CC 128
#define N2 (NN * NN)
#define KB 32

typedef __attribute__((ext_vector_type(2))) float v2f;
typedef __attribute__((ext_vector_type(4))) float v4f;
typedef __attribute__((ext_vector_type(8))) float v8f;
typedef __attribute__((ext_vector_type(4))) int   v4i;

__device__ __forceinline__ v8f wmma4(v2f a, v2f b, v8f c) {
  // D = A(16x4 f32) * B(4x16 f32) + C(16x16 f32)
  return __builtin_amdgcn_wmma_f32_16x16x4_f32(
      /*neg_a=*/false, a, /*neg_b=*/false, b,
      /*c_mod=*/(short)0, c, /*reuse_a=*/false, /*reuse_b=*/false);
}

__device__ __forceinline__ float sigm(float x) {
  return 1.0f / (1.0f + __expf(-x));
}

// ---- CDNA5 async memory->LDS path (ASYNCcnt) with sync fallback ------------
#if defined(__has_builtin) && __has_builtin(__builtin_amdgcn_global_load_async_to_lds_b128)
#define HAVE_ASYNC_LDS 1
#else
#define HAVE_ASYNC_LDS 0
#endif

#if HAVE_ASYNC_LDS
typedef __attribute__((address_space(1))) v4i* gv4i_p;   // global-side operand
typedef __attribute__((address_space(3))) v4i* lv4i_p;   // LDS-side operand
__device__ __forceinline__ void cp_async_b128(void* lds_dst, const float* gsrc) {
  // flat LDS address -> AS3: LDS offset is addr[31:0] (ISA 10.2 aperture rules)
  __builtin_amdgcn_global_load_async_to_lds_b128(
      (gv4i_p)(uintptr_t)gsrc, (lv4i_p)(uintptr_t)lds_dst, 0, 0);
}
template <int N> __device__ __forceinline__ void wait_asynccnt() {
#if defined(__has_builtin) && __has_builtin(__builtin_amdgcn_s_wait_asynccnt)
  __builtin_amdgcn_s_wait_asynccnt((short)N);
#else
  asm volatile("s_wait_asynccnt %0" ::"i"(N) : "memory");
#endif
}
#else
__device__ __forceinline__ void cp_async_b128(void* lds_dst, const float* gsrc) {
  *(v4f*)lds_dst = *(const v4f*)gsrc;
}
template <int N> __device__ __forceinline__ void wait_asynccnt() {}
#endif

// ---------------------------------------------------------------------------
// Kernel A: LayerNorm(act) -> proj/gate/glin GEMMs (WMMA f32), mask+sigmoid.
// Writes a_t, b_t in channel-major (C, N, N); g in (N^2, C).
// Block: 256 threads = 8 waves, handles 16 consecutive flattened positions.
// ---------------------------------------------------------------------------
__global__ __launch_bounds__(256) void kA(
    const float* __restrict__ act, const float* __restrict__ mask,
    const float* __restrict__ lnw, const float* __restrict__ lnb,
    const float* __restrict__ Wp, const float* __restrict__ Wg,
    const float* __restrict__ Wl,
    float* __restrict__ a_t, float* __restrict__ b_t, float* __restrict__ g)
{
  __shared__ float xs[16][132];   // 16 pos x 128 ch, pad -> stride 132 words
  __shared__ float mk[16];
  const int pos0 = blockIdx.x * 16;
  const int tid  = threadIdx.x;

  // ---- LayerNorm: 16 threads per position, 8 channels per thread ----
  {
    const int p = tid >> 4;
    const int l = tid & 15;
    const float* row = act + (size_t)(pos0 + p) * CC + l * 8;
    v4f t0 = *(const v4f*)(row);
    v4f t1 = *(const v4f*)(row + 4);
    float v[8] = {t0.x, t0.y, t0.z, t0.w, t1.x, t1.y, t1.z, t1.w};
    float s = 0.f, s2 = 0.f;
#pragma unroll
    for (int q = 0; q < 8; ++q) { s += v[q]; s2 += v[q] * v[q]; }
#pragma unroll
    for (int off = 8; off > 0; off >>= 1) {
      s  += __shfl_xor(s,  off, 16);
      s2 += __shfl_xor(s2, off, 16);
    }
    const float mean = s * (1.f / CC);
    const float var  = s2 * (1.f / CC) - mean * mean;
    const float inv  = rsqrtf(var + 1e-5f);
#pragma unroll
    for (int q = 0; q < 8; ++q) {
      const int c = l * 8 + q;
      xs[p][c] = (v[q] - mean) * inv * lnw[c] + lnb[c];
    }
    if (l == 0) mk[p] = mask[pos0 + p];
  }
  __syncthreads();

  // ---- WMMA GEMMs: M=16 positions, K=128 channels ----
  const int wave  = tid >> 5;
  const int lane  = tid & 31;
  const int lhalf = lane >> 4;
  const int l16   = lane & 15;

  const int dP0 = 32 * wave;        // proj/gate tile 0 output-channel base
  const int dP1 = 32 * wave + 16;   // proj/gate tile 1
  const int dL  = 16 * wave;        // glin tile

  v8f aP0 = {0,0,0,0,0,0,0,0};
  v8f aP1 = {0,0,0,0,0,0,0,0};
  v8f aG0 = {0,0,0,0,0,0,0,0};
  v8f aG1 = {0,0,0,0,0,0,0,0};
  v8f aL  = {0,0,0,0,0,0,0,0};

  // B fragment: B[k][n] = W[n_base + l16][k]; lanes<16 take K={k,k+1},
  // lanes>=16 take K={k+2,k+3}.
  const float* pP0 = Wp + (size_t)(dP0 + l16) * CC + 2 * lhalf;
  const float* pP1 = Wp + (size_t)(dP1 + l16) * CC + 2 * lhalf;
  const float* pG0 = Wg + (size_t)(dP0 + l16) * CC + 2 * lhalf;
  const float* pG1 = Wg + (size_t)(dP1 + l16) * CC + 2 * lhalf;
  const float* pL  = Wl + (size_t)(dL  + l16) * CC + 2 * lhalf;

#pragma unroll 4
  for (int k0 = 0; k0 < CC; k0 += 4) {
    v2f af;
    af.x = xs[l16][k0 + 2 * lhalf];
    af.y = xs[l16][k0 + 2 * lhalf + 1];
    v2f b0 = *(const v2f*)(pP0 + k0);
    v2f b1 = *(const v2f*)(pP1 + k0);
    v2f b2 = *(const v2f*)(pG0 + k0);
    v2f b3 = *(const v2f*)(pG1 + k0);
    v2f b4 = *(const v2f*)(pL  + k0);
    aP0 = wmma4(af, b0, aP0);
    aP1 = wmma4(af, b1, aP1);
    aG0 = wmma4(af, b2, aG0);
    aG1 = wmma4(af, b3, aG1);
    aL  = wmma4(af, b4, aL);
  }

  // ---- epilogue: p = proj * mask * sigmoid(gate); split a/b; g = sigmoid ----
  // waves 0..3 own d in [0,128) -> a ; waves 4..7 own d in [128,256) -> b
  float* base0 = (wave < 4) ? (a_t + (size_t)(dP0 + l16) * N2)
                            : (b_t + (size_t)(dP0 - CC + l16) * N2);
  float* base1 = (wave < 4) ? (a_t + (size_t)(dP1 + l16) * N2)
                            : (b_t + (size_t)(dP1 - CC + l16) * N2);
#pragma unroll
  for (int r = 0; r < 8; ++r) {
    const int m = r + 8 * lhalf;        // position within tile (C/D layout)
    const float mv = mk[m];
    base0[pos0 + m] = aP0[r] * mv * sigm(aG0[r]);
    base1[pos0 + m] = aP1[r] * mv * sigm(aG1[r]);
    g[(size_t)(pos0 + m) * CC + dL + l16] = sigm(aL[r]);
  }
}

// ---------------------------------------------------------------------------
// Kernel B: triangle einsum t[c,i,j] = sum_k a[c,i,k]*b[c,j,k]
// (128 batched 512x512x512 fp32 GEMMs, T = A * B^T). Block tile 64x128,
// 8 waves each own a 32x32 accumulator (2x2 WMMA tiles).
// K staged 32-wide through double-buffered LDS, filled by ASYNC-to-LDS
// copies (ASYNCcnt) overlapped with WMMA on the previous stage.
// Row stride 36 words: 16B-aligned for b128 LDS writes; 36*i mod 64 =
// 4*(9i mod 16) -> conflict-free fragment reads across 16 lanes.
// ---------------------------------------------------------------------------
__global__ __launch_bounds__(256) void kB(
    const float* __restrict__ a_t, const float* __restrict__ b_t,
    float* __restrict__ t_t)
{
  __shared__ float As[2][64][36];
  __shared__ float Bs[2][128][36];
  const int ch = blockIdx.z;
  const int i0 = blockIdx.y * 64;
  const int j0 = blockIdx.x * 128;
  const float* A = a_t + (size_t)ch * N2 + (size_t)i0 * NN;
  const float* B = b_t + (size_t)ch * N2 + (size_t)j0 * NN;

  const int tid   = threadIdx.x;
  const int wave  = tid >> 5;
  const int lane  = tid & 31;
  const int lhalf = lane >> 4;
  const int l16   = lane & 15;
  const int wm = (wave >> 2) * 32;    // 0 / 32
  const int wn = (wave & 3) * 32;     // 0 / 32 / 64 / 96
  const int lr = tid >> 2;            // 0..63 stage row
  const int c4 = (tid & 3) * 4;       // 0/4/8/12 stage col (x2 halves)

  v8f acc00 = {0,0,0,0,0,0,0,0};
  v8f acc01 = {0,0,0,0,0,0,0,0};
  v8f acc10 = {0,0,0,0,0,0,0,0};
  v8f acc11 = {0,0,0,0,0,0,0,0};

  // issue one K-stage (6 x b128 per thread: A 64x32, B 128x32)
  auto stage = [&](int buf, int k0) {
    cp_async_b128(&As[buf][lr][c4],           A + (size_t)lr * NN        + k0 + c4);
    cp_async_b128(&As[buf][lr][c4 + 16],      A + (size_t)lr * NN        + k0 + c4 + 16);
    cp_async_b128(&Bs[buf][lr][c4],           B + (size_t)lr * NN        + k0 + c4);
    cp_async_b128(&Bs[buf][lr][c4 + 16],      B + (size_t)lr * NN        + k0 + c4 + 16);
    cp_async_b128(&Bs[buf][lr + 64][c4],      B + (size_t)(lr + 64) * NN + k0 + c4);
    cp_async_b128(&Bs[buf][lr + 64][c4 + 16], B + (size_t)(lr + 64) * NN + k0 + c4 + 16);
  };

  stage(0, 0);
  for (int k0 = 0; k0 < NN; k0 += KB) {
    const int buf = (k0 / KB) & 1;
    __syncthreads();                      // buf^1 no longer being read by anyone
    if (k0 + KB < NN) {
      stage(buf ^ 1, k0 + KB);            // prefetch next stage (async)
      wait_asynccnt<6>();                 // current stage's 6 copies landed
    } else {
      wait_asynccnt<0>();
    }
    __syncthreads();                      // LDS visibility across waves

#pragma unroll
    for (int kk = 0; kk < KB; kk += 4) {
      const int ks = kk + 2 * lhalf;
      v2f a0, a1, b0v, b1v;
      a0.x  = As[buf][wm + l16][ks];       a0.y  = As[buf][wm + l16][ks + 1];
      a1.x  = As[buf][wm + 16 + l16][ks];  a1.y  = As[buf][wm + 16 + l16][ks + 1];
      b0v.x = Bs[buf][wn + l16][ks];       b0v.y = Bs[buf][wn + l16][ks + 1];
      b1v.x = Bs[buf][wn + 16 + l16][ks];  b1v.y = Bs[buf][wn + 16 + l16][ks + 1];
      acc00 = wmma4(a0, b0v, acc00);
      acc01 = wmma4(a0, b1v, acc01);
      acc10 = wmma4(a1, b0v, acc10);
      acc11 = wmma4(a1, b1v, acc11);
    }
  }

  float* T = t_t + (size_t)ch * N2;
#pragma unroll
  for (int r = 0; r < 8; ++r) {
    const int m = r + 8 * lhalf;
    T[(size_t)(i0 + wm +      m) * NN + j0 + wn +      l16] = acc00[r];
    T[(size_t)(i0 + wm +      m) * NN + j0 + wn + 16 + l16] = acc01[r];
    T[(size_t)(i0 + wm + 16 + m) * NN + j0 + wn +      l16] = acc10[r];
    T[(size_t)(i0 + wm + 16 + m) * NN + j0 + wn + 16 + l16] = acc11[r];
  }
}

// ---------------------------------------------------------------------------
// Kernel C: LayerNorm_c(t) -> out = (tn @ W_out^T) * g. 16 positions/block.
// ---------------------------------------------------------------------------
__global__ __launch_bounds__(256) void kC(
    const float* __restrict__ t_t,
    const float* __restrict__ lnw, const float* __restrict__ lnb,
    const float* __restrict__ Wo, const float* __restrict__ g,
    float* __restrict__ out)
{
  __shared__ float xs[16][132];
  const int pos0 = blockIdx.x * 16;
  const int tid  = threadIdx.x;

  // ---- transpose-load t tile: (C,N,N) -> xs[pos][c] ----
  {
    const int c  = tid >> 1;
    const int ph = (tid & 1) * 8;
    const float* src = t_t + (size_t)c * N2 + pos0 + ph;
    v4f v0 = *(const v4f*)(src);
    v4f v1 = *(const v4f*)(src + 4);
    xs[ph + 0][c] = v0.x; xs[ph + 1][c] = v0.y;
    xs[ph + 2][c] = v0.z; xs[ph + 3][c] = v0.w;
    xs[ph + 4][c] = v1.x; xs[ph + 5][c] = v1.y;
    xs[ph + 6][c] = v1.z; xs[ph + 7][c] = v1.w;
  }
  __syncthreads();

  // ---- LayerNorm over channels ----
  {
    const int p = tid >> 4;
    const int l = tid & 15;
    float v[8];
    float s = 0.f, s2 = 0.f;
#pragma unroll
    for (int q = 0; q < 8; ++q) {
      v[q] = xs[p][l * 8 + q];
      s += v[q]; s2 += v[q] * v[q];
    }
#pragma unroll
    for (int off = 8; off > 0; off >>= 1) {
      s  += __shfl_xor(s,  off, 16);
      s2 += __shfl_xor(s2, off, 16);
    }
    const float mean = s * (1.f / CC);
    const float var  = s2 * (1.f / CC) - mean * mean;
    const float inv  = rsqrtf(var + 1e-5f);
#pragma unroll
    for (int q = 0; q < 8; ++q) {
      const int c = l * 8 + q;
      xs[p][c] = (v[q] - mean) * inv * lnw[c] + lnb[c];
    }
  }
  __syncthreads();

  // ---- WMMA GEMM: M=16 pos, N=128 out channels (1 n-tile per wave), K=128 ----
  const int wave  = tid >> 5;
  const int lane  = tid & 31;
  const int lhalf = lane >> 4;
  const int l16   = lane & 15;
  const int d0    = 16 * wave;

  v8f acc = {0,0,0,0,0,0,0,0};
  const float* pW = Wo + (size_t)(d0 + l16) * CC + 2 * lhalf;
#pragma unroll 4
  for (int k0 = 0; k0 < CC; k0 += 4) {
    v2f af;
    af.x = xs[l16][k0 + 2 * lhalf];
    af.y = xs[l16][k0 + 2 * lhalf + 1];
    v2f bf = *(const v2f*)(pW + k0);
    acc = wmma4(af, bf, acc);
  }

#pragma unroll
  for (int r = 0; r < 8; ++r) {
    const int m = r + 8 * lhalf;
    const size_t o = (size_t)(pos0 + m) * CC + d0 + l16;
    out[o] = acc[r] * g[o];
  }
}

// ---------------------------------------------------------------------------
extern "C" void kernel_launch(void* const* d_in, const int* in_sizes, int n_in,
                              void* d_out, int out_size, void* d_ws, size_t ws_size,
                              hipStream_t stream) {
  (void)in_sizes; (void)n_in; (void)out_size; (void)ws_size;
  const float* act  = (const float*)d_in[0];
  const float* mask = (const float*)d_in[1];
  const float* lniw = (const float*)d_in[2];
  const float* lnib = (const float*)d_in[3];
  const float* Wp   = (const float*)d_in[4];
  const float* Wg   = (const float*)d_in[5];
  const float* lncw = (const float*)d_in[6];
  const float* lncb = (const float*)d_in[7];
  const float* Wo   = (const float*)d_in[8];
  const float* Wl   = (const float*)d_in[9];
  float* out = (float*)d_out;

  const size_t plane = (size_t)CC * N2;    // 33,554,432 floats = 128 MiB
  float* a_t = (float*)d_ws;               // (C,N,N)
  float* b_t = a_t + plane;                // (C,N,N)
  float* g   = b_t + plane;                // (N^2,C)
  float* t_t = g + plane;                  // (C,N,N)

  kA<<<N2 / 16, 256, 0, stream>>>(act, mask, lniw, lnib, Wp, Wg, Wl, a_t, b_t, g);
  kB<<<dim3(NN / 128, NN / 64, CC), 256, 0, stream>>>(a_t, b_t, t_t);
  kC<<<N2 / 16, 256, 0, stream>>>(t_t, lncw, lncb, Wo, g, out);
}